// SharedMultiCategoricalEncoder_9938554322950
// MI455X (gfx1250) — compile-verified
//
#include <hip/hip_runtime.h>

// Embedding-bag mean over hashed categorical ids.
//   x:   [B, C, L] int ids (0 = padding)        B*C = 262144, L = 8
//   emb: [9311, 64] fp32 (row 0 is zeros)
//   out: [B, C, 64] fp32 = sum(emb[x%9311] * (x>0)) / max(count, 1)
// Memory-bound gather: table (2.3 MiB) lives in L2/WGP$; output (64 MiB) is
// streamed with non-temporal b128 stores; per-block index tile is staged via
// the gfx1250 async global->LDS path (ASYNCcnt).

#ifndef NUM_HASH_BUCKETS
#define NUM_HASH_BUCKETS 9311
#endif

constexpr int kD            = 64;                        // OUT_CHANNELS
constexpr int kL            = 8;                         // bag length
constexpr int kThreads      = 256;                       // 8 waves (wave32)
constexpr int kLanesPerPair = 16;                        // 16 lanes x float4 = 64 floats
constexpr int kPairsPerBlock = kThreads / kLanesPerPair; // 16
constexpr int kIdxPerBlock   = kPairsPerBlock * kL;      // 128 indices / block

typedef float v4f __attribute__((ext_vector_type(4)));
typedef int   v4i __attribute__((ext_vector_type(4)));

// Address-space-qualified int4 pointers for the async-copy builtin
// (param types per clang diagnostic: int4 in AS1 / AS3).
typedef __attribute__((address_space(1))) v4i gas_v4i;
typedef __attribute__((address_space(3))) v4i las_v4i;

__global__ __launch_bounds__(kThreads)
void embag_mean_kernel(const int* __restrict__ x,
                       const float* __restrict__ emb,
                       float* __restrict__ out,
                       int npairs)
{
    __shared__ int s_idx[kIdxPerBlock];
    __shared__ int s_code[kIdxPerBlock];

    const int tid = threadIdx.x;
    const long long pair0    = (long long)blockIdx.x * kPairsPerBlock;
    const long long idx0     = pair0 * kL;
    const long long idxTotal = (long long)npairs * kL;

    // ---- Stage 1: async global->LDS copy of this block's 128 indices ----
    // One wave issues 32x b128 async copies (tracked on ASYNCcnt).
    if (tid < kIdxPerBlock / 4) {
        const long long g = idx0 + (long long)tid * 4;
        if (g + 3 < idxTotal) {
            const int* gp = x + g;
#if __has_builtin(__builtin_amdgcn_global_load_async_to_lds_b128)
            __builtin_amdgcn_global_load_async_to_lds_b128(
                (gas_v4i*)(v4i*)gp,
                (las_v4i*)(v4i*)&s_idx[tid * 4],
                /*offset=*/0, /*cpol=*/0);
#else
            *(v4i*)&s_idx[tid * 4] = *(const v4i*)gp;  // fallback: sync copy
#endif
        }
    }
#if __has_builtin(__builtin_amdgcn_s_wait_asynccnt)
    __builtin_amdgcn_s_wait_asynccnt(0);
#else
    asm volatile("s_wait_asynccnt 0" ::: "memory");
#endif
    __syncthreads();

    // ---- Stage 2: relu + hash(mod 9311) + padding mask, one lane/index ----
    if (tid < kIdxPerBlock) {
        int v = s_idx[tid];
        v = (v > 0) ? v : 0;                                   // relu
        unsigned h = (unsigned)v % (unsigned)NUM_HASH_BUCKETS; // magic-mul mod
        s_code[tid] = (v > 0) ? (int)h : -1;                   // -1 => padding
    }
    __syncthreads();

    // ---- Stage 3: gather-accumulate; 16 lanes x float4 per (b,c) pair ----
    const int group = tid >> 4;      // pair within block
    const int l16   = tid & 15;      // lane within pair
    const long long pair = pair0 + group;
    if (pair >= npairs) return;

    v4f  acc = {0.f, 0.f, 0.f, 0.f};
    float cnt = 0.f;
#pragma unroll
    for (int j = 0; j < kL; ++j) {
        const int c = s_code[group * kL + j];  // uniform across the 16 lanes
        if (c >= 0) {
            const v4f* row = (const v4f*)(emb + (size_t)c * kD);
            acc += row[l16];                   // 256B coalesced b128 gather
            cnt += 1.0f;
        }
    }
    const float inv = 1.0f / fmaxf(cnt, 1.0f);
    acc *= inv;

    v4f* op = (v4f*)(out + (size_t)pair * kD + (size_t)l16 * 4);
    __builtin_nontemporal_store(acc, op);      // streaming 64MB output: TH=NT
}

extern "C" void kernel_launch(void* const* d_in, const int* in_sizes, int n_in,
                              void* d_out, int out_size, void* d_ws, size_t ws_size,
                              hipStream_t stream) {
    const int*   x   = (const int*)d_in[0];     // [B, C, L] ids
    const float* emb = (const float*)d_in[1];   // [9311, 64]
    float*       out = (float*)d_out;           // [B, C, 64]

    const int npairs = in_sizes[0] / kL;        // B*C = 262144
    const int blocks = (npairs + kPairsPerBlock - 1) / kPairsPerBlock;
    embag_mean_kernel<<<blocks, kThreads, 0, stream>>>(x, emb, out, npairs);
}